// GeodesicDistanceMap_21672404975878
// MI455X (gfx1250) — compile-verified
//
#include <hip/hip_runtime.h>
#include <math.h>

// Geodesic log-map on S^44: out = (x - cos_t * e1) * theta/sin(theta)
// B=2, X=Y=Z=96, C=45  -> 1,769,472 voxels, 79,626,240 floats in/out.
// Pure streaming op: stage tiles via CDNA5 async global->LDS DMA, compute
// per-voxel scalars in LDS, write back with nontemporal b128 stores.

#define CCH 45
#define VOX_PER_BLOCK 256
#define THREADS 256
#define TILE_ELEMS (VOX_PER_BLOCK * CCH) /* 11520 floats = 45 KiB */

typedef __attribute__((ext_vector_type(4))) float f32x4;
typedef int vint4 __attribute__((vector_size(4 * sizeof(int))));

typedef __attribute__((address_space(3))) vint4 s_vint4;
typedef __attribute__((address_space(3))) int   s_int;

// Low 32 bits of a generic pointer into LDS are the AS(3) byte offset
__device__ static inline s_vint4* to_lds4(void* p) {
  return (s_vint4*)(unsigned)(size_t)p;
}
__device__ static inline s_int* to_lds1(void* p) {
  return (s_int*)(unsigned)(size_t)p;
}

__device__ static inline void async_copy_b128(const float* g, float* l) {
#if __has_builtin(__builtin_amdgcn_global_load_async_to_lds_b128)
  __builtin_amdgcn_global_load_async_to_lds_b128((vint4*)g, to_lds4(l), 0, 0);
#else
#warning "async b128 builtin missing: using inline asm global_load_async_to_lds_b128"
  unsigned lds_addr = (unsigned)(size_t)l;
  unsigned long long gaddr = (unsigned long long)(size_t)g;
  asm volatile("global_load_async_to_lds_b128 %0, %1, off"
               :: "v"(lds_addr), "v"(gaddr) : "memory");
#endif
}

__device__ static inline void async_copy_b32(const float* g, float* l) {
#if __has_builtin(__builtin_amdgcn_global_load_async_to_lds_b32)
  __builtin_amdgcn_global_load_async_to_lds_b32((int*)g, to_lds1(l), 0, 0);
#else
  unsigned lds_addr = (unsigned)(size_t)l;
  unsigned long long gaddr = (unsigned long long)(size_t)g;
  asm volatile("global_load_async_to_lds_b32 %0, %1, off"
               :: "v"(lds_addr), "v"(gaddr) : "memory");
#endif
}

__device__ static inline void wait_async0() {
#if __has_builtin(__builtin_amdgcn_s_wait_asynccnt)
  __builtin_amdgcn_s_wait_asynccnt(0);
#else
#warning "s_wait_asynccnt builtin missing: using inline asm"
  asm volatile("s_wait_asynccnt 0" ::: "memory");
#endif
}

__global__ __launch_bounds__(THREADS)
void geo_logmap_kernel(const float* __restrict__ in, float* __restrict__ out,
                       long long total) {
  __shared__ float tile[TILE_ELEMS];
  __shared__ float cosA[VOX_PER_BLOCK];
  __shared__ float scaleA[VOX_PER_BLOCK];

  const int tid = threadIdx.x;
  const long long base = (long long)blockIdx.x * TILE_ELEMS;
  if (base >= total) return;
  const long long left = total - base;
  const int elems = (left < (long long)TILE_ELEMS) ? (int)left : TILE_ELEMS;
  const int nvec  = elems >> 2;       // b128 chunks
  const int rem   = elems & 3;        // scalar tail (only possible on last block)
  const int nvox  = elems / CCH;

  const float* gin  = in  + base;
  float*       gout = out + base;

  // ---- Phase 1: async DMA tile into LDS (b128, fully coalesced) ----
  for (int k = tid; k < nvec; k += THREADS)
    async_copy_b128(gin + 4 * k, tile + 4 * k);
  if (tid < rem)
    async_copy_b32(gin + 4 * nvec + tid, tile + 4 * nvec + tid);
  wait_async0();          // each wave drains its own ASYNCcnt ...
  __syncthreads();        // ... then the workgroup barrier makes LDS visible

  // ---- Phase 2: per-voxel scalars (stride-45 LDS reads: bank-conflict-free) ----
  if (tid < nvox) {
    const float eps = 1e-6f;
    float c = tile[tid * CCH + 1];
    c = fminf(fmaxf(c, -1.0f + eps), 1.0f - eps);
    float theta = acosf(c);
    float s = sqrtf(fmaxf(1.0f - c * c, 1e-30f)); // == sin(theta), theta in (0,pi)
    cosA[tid]   = c;
    scaleA[tid] = theta / s;
  }
  __syncthreads();

  // ---- Phase 3: elementwise map, nontemporal b128 stores ----
  for (int k = tid; k < nvec; k += THREADS) {
    f32x4 v = *(const f32x4*)(tile + 4 * k);
    const unsigned j = 4u * (unsigned)k;
    f32x4 r;
#pragma unroll
    for (int e = 0; e < 4; ++e) {
      unsigned idx = j + (unsigned)e;
      unsigned vox = idx / CCH;            // const-div -> mulhi
      unsigned ch  = idx - vox * CCH;
      float x  = v[e];
      float sc = scaleA[vox];
      r[e] = (ch == 1u ? (x - cosA[vox]) : x) * sc;
    }
    __builtin_nontemporal_store(r, (f32x4*)(gout + j));
  }
  if (tid < rem) {
    unsigned idx = (unsigned)(4 * nvec + tid);
    unsigned vox = idx / CCH;
    unsigned ch  = idx - vox * CCH;
    float x = tile[idx];
    float o = (ch == 1u ? (x - cosA[vox]) : x) * scaleA[vox];
    __builtin_nontemporal_store(o, gout + idx);
  }
}

extern "C" void kernel_launch(void* const* d_in, const int* in_sizes, int n_in,
                              void* d_out, int out_size, void* d_ws, size_t ws_size,
                              hipStream_t stream) {
  (void)in_sizes; (void)n_in; (void)d_ws; (void)ws_size;
  const float* in = (const float*)d_in[0];
  float* out = (float*)d_out;
  long long total = (long long)out_size;            // 79,626,240 floats
  long long nblk = (total + TILE_ELEMS - 1) / TILE_ELEMS;  // 6912 for ref shape
  geo_logmap_kernel<<<dim3((unsigned)nblk), dim3(THREADS), 0, stream>>>(in, out, total);
}